// AttentionRefinementModule_46153718562935
// MI455X (gfx1250) — compile-verified
//
#include <hip/hip_runtime.h>

// ---------------------------------------------------------------------------
// AttentionRefinementModule for MI455X (gfx1250, wave32).
//  - conv5x5 + 1x1 proj as implicit GEMM on V_WMMA_F32_16X16X4_F32 (fp32
//    tensor path: op is HBM-bound, fp32 costs nothing, keeps accuracy).
//  - input strip staged into LDS by the Tensor Data Mover (TENSOR_LOAD_TO_LDS,
//    padded-row descriptor, channel iteration), waited with s_wait_tensorcnt.
//  - K-loop fully unrolled: constexpr im2col offset table -> constant-offset
//    ds_load_b32 gathers, constant-offset global_load_b64 weight streams.
// ---------------------------------------------------------------------------

typedef __attribute__((ext_vector_type(2))) float v2f;
typedef __attribute__((ext_vector_type(8))) float v8f;
typedef unsigned int u32x4 __attribute__((ext_vector_type(4)));
typedef int i32x4 __attribute__((ext_vector_type(4)));
typedef int i32x8 __attribute__((ext_vector_type(8)));

static constexpr int NH   = 8;        // heads
static constexpr int C2   = 16;       // 2*NH conv input channels
static constexpr int DCH  = 32;       // conv output channels
static constexpr int T    = 128;
static constexpr int H    = 32;
static constexpr int W    = 64;
static constexpr int L    = H * W;    // 2048
static constexpr int KTOT = C2 * 25;  // 400 = K dim of conv GEMM
static constexpr int NKC  = KTOT / 4; // 100 k-chunks of 4
static constexpr int ROWP = 68;       // padded LDS row pitch (dwords)
static constexpr float BN_EPS = 1e-5f;

// compile-time im2col LDS offsets (dwords), lo = lane 0..15 (K+0/1),
// hi = lane 16..31 (K+2/3), relative to &sIn[0][yloc][x0+nIdx]
struct OffTab { int lo[NKC][2]; int hi[NKC][2]; };
static constexpr OffTab makeOff() {
    OffTab t{};
    for (int kc = 0; kc < NKC; ++kc)
        for (int v = 0; v < 2; ++v) {
            int k0 = kc * 4 + v, k1 = kc * 4 + 2 + v;
            t.lo[kc][v] = (k0 / 25) * (6 * ROWP) + ((k0 % 25) / 5) * ROWP + (k0 % 5);
            t.hi[kc][v] = (k1 / 25) * (6 * ROWP) + ((k1 % 25) / 5) * ROWP + (k1 % 5);
        }
    return t;
}
static constexpr OffTab OFFT = makeOff();

#define RFL(x) __builtin_amdgcn_readfirstlane((int)(x))

// ---------------------------------------------------------------------------
// Kernel 1: exclusive cumsum over t, concat [prev;curr] -> 16 channels,
// written directly in (b*t, c, h*w) layout. Coalesced across l.
// ---------------------------------------------------------------------------
__global__ __launch_bounds__(256) void k_cumsum(const float* __restrict__ prev,
                                                const float* __restrict__ curr,
                                                float* __restrict__ A, int b) {
    int bc = blockIdx.y;
    int bb = bc / C2, c = bc % C2;
    int li = blockIdx.x * 256 + threadIdx.x;
    int head = c & 7;
    const float* src = (c < 8) ? prev : curr;
    size_t sbase = ((size_t)(bb * NH + head) * T) * L + li;
    float run = 0.f;
    for (int t = 0; t < T; ++t) {
        A[(((size_t)(bb * T + t)) * C2 + c) * L + li] = run;  // exclusive
        run += src[sbase + (size_t)t * L];
    }
}

// ---------------------------------------------------------------------------
// Kernel 2a: pack weights into WMMA fragment lane layouts.
// B 4x16 fragment (mirror of documented A 16x4): VGPR v, lane L:
//   K = 2*(L>=16)+v, N = L%16.
// ---------------------------------------------------------------------------
__global__ void k_pack(const float* __restrict__ conv_w,  // (32,16,5,5)=oc*400+k
                       const float* __restrict__ proj_w,  // (8,32)
                       float* __restrict__ Wp, float* __restrict__ Pp) {
    for (int i = threadIdx.x; i < NKC * 2 * 32 * 2; i += 256) {
        int v = i & 1, lane = (i >> 1) & 31, half = (i >> 6) & 1, kc = i >> 7;
        int k = 4 * kc + ((lane >= 16) ? 2 : 0) + v;
        int oc = half * 16 + (lane & 15);
        Wp[i] = (k < KTOT) ? conv_w[oc * KTOT + k] : 0.f;
    }
    for (int j = threadIdx.x; j < 8 * 32 * 2; j += 256) {
        int v = j & 1, lane = (j >> 1) & 31, kc = j >> 6;
        int head = lane & 15;
        int k = 4 * kc + ((lane >= 16) ? 2 : 0) + v;
        Pp[j] = (head < NH && k < DCH) ? proj_w[head * DCH + k] : 0.f;
    }
}

// ---------------------------------------------------------------------------
// TDM: async-load one wave's 2-channel strip into LDS.
//  tile: 64 cols x tileRows rows, iterate 2 channels.
//  LDS pitch 68 dwords built with pad_interval=64dw, pad_amount=4dw;
//  lds start at +2 dwords gives the left halo; bottom halo rows are
//  auto-zero-filled via tensor_dim1 OOB; top halo/pad cols pre-zeroed.
// ---------------------------------------------------------------------------
static __device__ __forceinline__ void tdm_load_strip(const float* gsrc,
                                                      unsigned int ldsByte,
                                                      int tileRows, int tensorRows) {
    unsigned long long ga = (unsigned long long)(uintptr_t)gsrc;
    u32x4 g0;
    g0[0] = (unsigned)RFL(1);                                   // count=1 (user D#)
    g0[1] = (unsigned)RFL(ldsByte);                             // lds_addr
    g0[2] = (unsigned)RFL((unsigned int)ga);                    // global_addr lo
    g0[3] = (unsigned)RFL(((unsigned int)(ga >> 32) & 0x01FFFFFFu) | 0x80000000u); // hi|type=2

    i32x8 g1;
    g1[0] = RFL((2 << 16) | (1 << 19) | (1 << 20) | (5 << 22) | (3 << 25));
    //        data_size=4B  iterate     pad_en      pad@64dw     pad=4dw
    g1[1] = RFL((W & 0xFFFF) << 16);                            // tensor_dim0 lo16
    g1[2] = RFL((tensorRows & 0xFFFF) << 16);                   // dim0 hi=0 | dim1 lo16
    g1[3] = RFL(((tensorRows >> 16) & 0xFFFF) | (W << 16));     // dim1 hi | tile_dim0=64
    g1[4] = RFL(tileRows & 0xFFFF);                             // tile_dim1 | tile_dim2=0
    g1[5] = RFL(W);                                             // tensor_dim0_stride=64
    g1[6] = RFL(0);
    g1[7] = RFL(0);

    i32x4 g2;
    g2[0] = RFL(C2);                                            // tensor_dim2 (inert, 2D tile)
    g2[1] = RFL(6 * ROWP);                                      // lds_addr_increment (elems)
    g2[2] = RFL(H * W);                                         // global_addr_increment (elems)
    g2[3] = RFL(1 << 16);                                       // iterate_count=1 -> 2 iters

    i32x4 g3;
    g3[0] = RFL(0); g3[1] = RFL(0); g3[2] = RFL(0); g3[3] = RFL(0);

#if __clang_major__ >= 23
    i32x8 g4;
    g4[0] = 0; g4[1] = 0; g4[2] = 0; g4[3] = 0;
    g4[4] = 0; g4[5] = 0; g4[6] = 0; g4[7] = 0;
    __builtin_amdgcn_tensor_load_to_lds(g0, g1, g2, g3, g4, 0);
#else
    __builtin_amdgcn_tensor_load_to_lds(g0, g1, g2, g3, 0);
#endif
}

// ---------------------------------------------------------------------------
// Kernel 2: conv5x5 + bias + relu + mask + 1x1 proj, fused, fp32 WMMA.
// One workgroup (8 waves) per (image bt, 2-row group); one 16-pixel tile per
// wave: 200 conv WMMAs + 8 proj WMMAs, K-loop fully unrolled.
// ---------------------------------------------------------------------------
__global__ __launch_bounds__(256) void k_conv(const float* __restrict__ A,
                                              const float* __restrict__ Wp,
                                              const float* __restrict__ Pp,
                                              const float* __restrict__ conv_b,
                                              const int* __restrict__ kpm,
                                              float* __restrict__ proj, int b) {
    __shared__ __align__(16) float sIn[C2][6][ROWP];  // 2 rows + halo, pitch 68
    __shared__ float sCov[8][16][DCH];                // per-wave cov bounce

    const int bt = blockIdx.y;
    const int y0 = blockIdx.x * 2;
    const int bb = bt / T;
    const int tid = threadIdx.x;
    const int wave = tid >> 5, lane = tid & 31;
    const int laneHi = (lane >= 16) ? 1 : 0;
    const int nIdx = lane & 15;

    __builtin_prefetch(Wp, 0, 3);                     // warm packed weights (L2)

    // pre-zero LDS strip: provides halo cols + top halo rows (TDM zero-fills
    // bottom OOB rows itself, then overwrites the interior)
    {
        float4 z4 = make_float4(0.f, 0.f, 0.f, 0.f);
        float4* z = (float4*)&sIn[0][0][0];
        for (int i = tid; i < C2 * 6 * ROWP / 4; i += 256) z[i] = z4;
    }
    __syncthreads();

    // each wave DMAs its 2 channels (wave-uniform descriptor, EXEC-independent)
    {
        int clipTop = (y0 == 0) ? 2 : 0;
        int grow0 = y0 - 2 + clipTop;                 // first valid source row
        int tileRows = 6 - clipTop;
        int tensorRows = H - grow0;                   // bottom OOB -> zero fill
        const float* gsrc = A + ((size_t)bt * C2 + wave * 2) * L + (size_t)grow0 * W;
        unsigned int ldsByte =
            (unsigned int)(uintptr_t)&sIn[wave * 2][clipTop][2];
        tdm_load_strip(gsrc, ldsByte, tileRows, tensorRows);
    }
    __builtin_amdgcn_s_wait_tensorcnt(0);
    __syncthreads();

    const int yloc = wave >> 2;                       // row within 2-row group
    const int y = y0 + yloc;
    const int x0 = (wave & 3) * 16;

    const float* pA = &sIn[0][yloc][x0 + nIdx];       // per-lane gather base
    const v2f* wpl = (const v2f*)Wp + lane;           // per-lane weight base
    const v2f* ppl = (const v2f*)Pp + lane;

    v8f acc0 = {0.f,0.f,0.f,0.f,0.f,0.f,0.f,0.f};
    v8f acc1 = {0.f,0.f,0.f,0.f,0.f,0.f,0.f,0.f};

    // conv GEMM: M=16 pixels, N=2x16 oc, K=400; fully unrolled so the im2col
    // gathers and weight streams become constant-offset ds/global loads
    #pragma unroll
    for (int kc = 0; kc < NKC; ++kc) {
        v2f a;
        #pragma unroll
        for (int v = 0; v < 2; ++v)
            a[v] = pA[laneHi ? OFFT.hi[kc][v] : OFFT.lo[kc][v]];
        v2f b0 = wpl[kc * 64];                        // oc 0..15
        v2f b1 = wpl[kc * 64 + 32];                   // oc 16..31
        acc0 = __builtin_amdgcn_wmma_f32_16x16x4_f32(false, a, false, b0,
                                                     (short)0, acc0, false, false);
        acc1 = __builtin_amdgcn_wmma_f32_16x16x4_f32(false, a, false, b1,
                                                     (short)0, acc1, false, false);
    }

    // epilogue: bias + relu + spatial mask; bounce cov tile through LDS
    // D layout: lane -> N (oc), VGPR r -> M = r + 8*laneHi (pixel)
    {
        float bias0 = conv_b[nIdx], bias1 = conv_b[16 + nIdx];
        #pragma unroll
        for (int r = 0; r < 8; ++r) {
            int m = r + 8 * laneHi;
            int x = x0 + m;
            int msk = kpm[bb * L + y * W + x];
            sCov[wave][m][nIdx]      = msk ? 0.f : fmaxf(acc0[r] + bias0, 0.f);
            sCov[wave][m][16 + nIdx] = msk ? 0.f : fmaxf(acc1[r] + bias1, 0.f);
        }
    }
    __syncthreads();

    // 1x1 proj: D'(head x pixel) = Pw(16x32, 8 used) @ cov^T(32 x 16px)
    v8f p = {0.f,0.f,0.f,0.f,0.f,0.f,0.f,0.f};
    #pragma unroll
    for (int kc = 0; kc < 8; ++kc) {
        v2f a = ppl[kc * 32];
        v2f bf;
        #pragma unroll
        for (int v = 0; v < 2; ++v) {
            int ock = kc * 4 + 2 * laneHi + v;
            bf[v] = sCov[wave][nIdx][ock];            // B: lane -> pixel N, K=oc
        }
        p = __builtin_amdgcn_wmma_f32_16x16x4_f32(false, a, false, bf,
                                                  (short)0, p, false, false);
    }
    // D': lane -> pixel N, VGPR r -> head M (lanes 0..15 hold heads 0..7)
    if (lane < 16) {
        size_t basep = ((size_t)bt * NH) * L + y * W + x0 + lane;
        #pragma unroll
        for (int r = 0; r < NH; ++r) proj[basep + (size_t)r * L] = p[r];
    }
}

// ---------------------------------------------------------------------------
// Kernel 3: deterministic BN stats. proj is exactly 0 at masked pixels, so
// plain sum/sumsq equal the masked sums; only cnt needs the mask.
// ---------------------------------------------------------------------------
__global__ __launch_bounds__(256) void k_stats(const float* __restrict__ proj,
                                               const int* __restrict__ kpm,
                                               double* __restrict__ stats, int b) {
    __shared__ double red[256], red2[256];
    const int tid = threadIdx.x, blk = blockIdx.x;
    if (blk < NH) {
        long tot = (long)b * T * L;
        double s = 0.0, s2 = 0.0;
        for (long i = tid; i < tot; i += 256) {
            long bt = i / L, hw = i - bt * L;
            float x = proj[(bt * NH + blk) * L + hw];
            s += x; s2 += (double)x * (double)x;
        }
        red[tid] = s; red2[tid] = s2;
        __syncthreads();
        for (int st = 128; st > 0; st >>= 1) {
            if (tid < st) { red[tid] += red[tid + st]; red2[tid] += red2[tid + st]; }
            __syncthreads();
        }
        if (tid == 0) { stats[blk * 2] = red[0]; stats[blk * 2 + 1] = red2[0]; }
    } else {
        long cnt = 0;
        int tot = b * L;
        for (int i = tid; i < tot; i += 256) cnt += (kpm[i] == 0);
        red[tid] = (double)cnt;
        __syncthreads();
        for (int st = 128; st > 0; st >>= 1) {
            if (tid < st) red[tid] += red[tid + st];
            __syncthreads();
        }
        if (tid == 0) { double c = red[0] * (double)T; stats[16] = (c > 1.0) ? c : 1.0; }
    }
}

// ---------------------------------------------------------------------------
// Kernel 4: BN apply + relayout (b t) n hw -> (b n) t hw. Coalesced on hw.
// ---------------------------------------------------------------------------
__global__ __launch_bounds__(256) void k_apply(const float* __restrict__ proj,
                                               const int* __restrict__ kpm,
                                               const double* __restrict__ stats,
                                               const float* __restrict__ gamma,
                                               const float* __restrict__ beta,
                                               float* __restrict__ out, long total) {
    long idx = (long)blockIdx.x * 256 + threadIdx.x;
    if (idx >= total) return;
    long hw = idx % L;
    long r1 = idx / L;
    long t  = r1 % T;
    long r2 = r1 / T;
    long n1 = r2 % NH;
    long bb = r2 / NH;
    float x = proj[(((size_t)(bb * T + t)) * NH + n1) * L + hw];
    if (kpm[bb * L + hw]) {
        out[idx] = x;                                 // masked: pre-BN (== 0)
    } else {
        float cnt  = (float)stats[16];
        float mean = (float)stats[2 * n1] / cnt;
        float var  = (float)stats[2 * n1 + 1] / cnt - mean * mean;
        out[idx] = gamma[n1] * (x - mean) * rsqrtf(var + BN_EPS) + beta[n1];
    }
}

// ---------------------------------------------------------------------------
// Launcher. ws: [cumsum (b*t,16,L) f32 | proj (b*t,8,L) f32 |
//                Wp 12800 f32 | Pp 512 f32 | stats 17 f64]  ~= 201 MB
// ---------------------------------------------------------------------------
extern "C" void kernel_launch(void* const* d_in, const int* in_sizes, int n_in,
                              void* d_out, int out_size, void* d_ws, size_t ws_size,
                              hipStream_t stream) {
    const float* prev   = (const float*)d_in[0];
    const float* curr   = (const float*)d_in[1];
    const int*   kpm    = (const int*)d_in[2];        // True = padded
    const float* conv_w = (const float*)d_in[4];
    const float* conv_b = (const float*)d_in[5];
    const float* proj_w = (const float*)d_in[6];
    const float* gamma  = (const float*)d_in[7];
    const float* beta   = (const float*)d_in[8];
    float* out = (float*)d_out;

    int b = in_sizes[2] / L;                          // mask is (b, L)

    char* ws = (char*)d_ws;
    size_t szA = (size_t)b * T * C2 * L * sizeof(float);
    size_t szP = (size_t)b * T * NH * L * sizeof(float);
    float*  A     = (float*)ws;
    float*  proj  = (float*)(ws + szA);
    float*  Wp    = (float*)(ws + szA + szP);
    float*  Pp    = Wp + NKC * 2 * 32 * 2;
    double* stats = (double*)(ws + szA + szP + (NKC * 2 * 32 * 2 + 512) * sizeof(float));

    dim3 g1(L / 256, b * C2);
    k_cumsum<<<g1, 256, 0, stream>>>(prev, curr, A, b);

    k_pack<<<1, 256, 0, stream>>>(conv_w, proj_w, Wp, Pp);

    dim3 g2(H / 2, b * T);
    k_conv<<<g2, 256, 0, stream>>>(A, Wp, Pp, conv_b, kpm, proj, b);

    k_stats<<<NH + 1, 256, 0, stream>>>(proj, kpm, stats, b);

    long total = (long)b * NH * T * L;
    k_apply<<<(unsigned)((total + 255) / 256), 256, 0, stream>>>(proj, kpm, stats,
                                                                 gamma, beta, out, total);
}